// SelfSupervisedContrastiveLoss_62113817035247
// MI455X (gfx1250) — compile-verified
//
#include <hip/hip_runtime.h>
#include <math.h>

// Self-supervised contrastive loss, fused for MI455X (gfx1250).
//   Stage 1: row-normalize f, f_pos (f32), zero accumulators.
//   Stage 2: fused Gram/exp/reduce using V_WMMA_F32_16X16X4_F32 (exact f32 path).
//            e_ff / e_pp are symmetric: only tiles tj>=ti are computed; a tile's
//            column sums are scattered as the mirrored tile's row sums (33% fewer
//            WMMA tiles overall). Row/col reductions use single-instruction
//            DS_SWIZZLE_B32 xor butterflies instead of guarded ds_bpermute.
//   Stage 3: loss = mean_i [ 0.5*(log d1 + log d2) - log pos ].

typedef float v2f __attribute__((ext_vector_type(2)));
typedef float v8f __attribute__((ext_vector_type(8)));

#define B_N 4096
#define D_K 128
#define T_TILES 256                     // 4096/16 tile rows
#define TRI_TILES 32896                 // T*(T+1)/2
#define FP_TILES 65536                  // T*T
#define TOTAL_WAVES (FP_TILES + 2 * TRI_TILES)   // 131328, divisible by 8

// One-instruction lane butterflies via ds_swizzle (group-of-32 xor mode).
#define SWZ_ADD(s, imm) \
  (s) += __int_as_float(__builtin_amdgcn_ds_swizzle(__float_as_int(s), (imm)))

__global__ void ssc_normalize(const float* __restrict__ f,
                              const float* __restrict__ fpos,
                              float* __restrict__ fn, float* __restrict__ pn,
                              float* __restrict__ acc1, float* __restrict__ acc2) {
  __shared__ float red[128];
  const int row = blockIdx.x;       // 0..2*B_N-1
  const int tid = threadIdx.x;      // 0..127
  const float* src = (row < B_N) ? f : fpos;
  float* dst       = (row < B_N) ? fn : pn;
  const int r = row & (B_N - 1);
  float v = src[r * D_K + tid];
  red[tid] = v * v;
  __syncthreads();
  for (int off = 64; off > 0; off >>= 1) {
    if (tid < off) red[tid] += red[tid + off];
    __syncthreads();
  }
  const float norm  = sqrtf(red[0]);
  const float scale = 1.0f / fmaxf(norm, 1e-8f);
  dst[r * D_K + tid] = v * scale;
  // zero the global accumulators (ws is not re-initialized between calls)
  if (row < B_N && tid == 0) { acc1[r] = 0.0f; acc2[r] = 0.0f; }
}

__device__ __forceinline__ int tri_base(int r) {       // #tiles in rows < r
  return r * T_TILES - (r * (r - 1)) / 2;
}

// One wave computes one 16x16 tile.
// mat 1 (fn@pn^T, full):      rowsums->acc1, colsums->acc2, diag->pos
// mat 0 (fn@fn^T, tj>=ti):    rowsums->acc1; if tj>ti colsums->acc1 (mirror rows)
// mat 2 (pn@pn^T, tj>=ti):    rowsums->acc2; if tj>ti colsums->acc2 (mirror rows)
__global__ void __launch_bounds__(256) ssc_gram(
    const float* __restrict__ fn, const float* __restrict__ pn,
    float* __restrict__ pos, float* __restrict__ acc1, float* __restrict__ acc2) {
  const int lane = threadIdx.x & 31;
  const int wave = threadIdx.x >> 5;
  const int w    = blockIdx.x * 8 + wave;   // 0 .. TOTAL_WAVES-1

  int mat, ti, tj;
  if (w < FP_TILES) {
    mat = 1;
    ti  = w >> 8;
    tj  = w & 255;
  } else {
    int t = w - FP_TILES;
    mat = (t < TRI_TILES) ? 0 : 2;
    if (t >= TRI_TILES) t -= TRI_TILES;
    // triangular decode: largest ti with tri_base(ti) <= t  (sqrt estimate + fixup)
    int r = (int)floorf((513.0f - sqrtf(263169.0f - 8.0f * (float)t)) * 0.5f);
    if (r < 0) r = 0;
    if (r > 255) r = 255;
    while (r < 255 && tri_base(r + 1) <= t) ++r;
    while (r > 0 && tri_base(r) > t) --r;
    ti = r;
    tj = ti + (t - tri_base(ti));
  }
  const int i0 = ti << 4;
  const int j0 = tj << 4;

  const float* X = (mat == 2) ? pn : fn;
  const float* Y = (mat == 0) ? fn : pn;

  // f32 WMMA 16x16x4 operand layout (ISA 7.12.2):
  //  A (16x4): lane l -> row (l&15), K = 2*(l>>4) + {0,1}
  //  B (4x16): lane l -> col (l&15), same K pattern -> identical per-lane load from Y.
  const int mrow = lane & 15;
  const int kk   = (lane >> 4) << 1;
  const float* xp = X + (i0 + mrow) * D_K + kk;
  const float* yp = Y + (j0 + mrow) * D_K + kk;

  v8f c = {};
#pragma unroll 8
  for (int k = 0; k < D_K; k += 4) {
    v2f a = *(const v2f*)(xp + k);
    v2f b = *(const v2f*)(yp + k);
    c = __builtin_amdgcn_wmma_f32_16x16x4_f32(
        /*neg_a=*/false, a, /*neg_b=*/false, b,
        /*c_mod=*/(short)0, c, /*reuse_a=*/false, /*reuse_b=*/false);
  }

  // C/D layout: VGPR r -> row (r or r+8); col = lane&15 in both halves.
  const int mbase = i0 + ((lane >> 4) << 3);
  const int n     = j0 + mrow;
  float* rowDst   = (mat == 2) ? acc2 : acc1;
  float* colDst   = (mat == 0) ? acc1 : acc2;
  const bool doCol = (mat == 1) || (ti != tj);   // symmetric diag tiles: rows only
  float colAcc    = 0.0f;

#pragma unroll
  for (int r = 0; r < 8; ++r) {
    const int  m    = mbase + r;
    const float e   = __expf(c[r]);              // TAU = 1
    const bool diag = (m == n);                  // only possible when ti==tj
    if (mat == 1 && diag) pos[m] = e;            // positive-pair similarity
    float val = diag ? 0.0f : e;
    if (doCol) colAcc += val;
    // half-wave (16-lane) row reduction: xor 1,2,4,8 stay within each half
    float s = val;
    SWZ_ADD(s, 0x041f);  // SWAPX1
    SWZ_ADD(s, 0x081f);  // SWAPX2
    SWZ_ADD(s, 0x101f);  // SWAPX4
    SWZ_ADD(s, 0x201f);  // SWAPX8
    if (mrow == 0) atomicAdd(&rowDst[m], s);
  }
  if (doCol) {
    // column n lives in lanes l and l^16; combine halves, one atomic per column
    SWZ_ADD(colAcc, 0x401f);  // SWAPX16
    if (lane < 16) atomicAdd(&colDst[n], colAcc);
  }
}

__global__ void ssc_loss(const float* __restrict__ pos,
                         const float* __restrict__ acc1,
                         const float* __restrict__ acc2,
                         float* __restrict__ out) {
  __shared__ float red[256];
  const int tid = threadIdx.x;
  float sum = 0.0f;
  for (int i = tid; i < B_N; i += 256) {
    const float p  = pos[i];
    const float d1 = p + acc1[i];   // denom1 = pos + rowsum(e_ff) + rowsum(e_fp)
    const float d2 = p + acc2[i];   // denom2 = pos + colsum(e_fp) + rowsum(e_pp)
    sum += 0.5f * (__logf(d1) + __logf(d2)) - __logf(p);
  }
  red[tid] = sum;
  __syncthreads();
  for (int off = 128; off > 0; off >>= 1) {
    if (tid < off) red[tid] += red[tid + off];
    __syncthreads();
  }
  if (tid == 0) out[0] = red[0] / (float)B_N;
}

extern "C" void kernel_launch(void* const* d_in, const int* in_sizes, int n_in,
                              void* d_out, int out_size, void* d_ws, size_t ws_size,
                              hipStream_t stream) {
  const float* f    = (const float*)d_in[0];
  const float* fpos = (const float*)d_in[1];

  float* fn   = (float*)d_ws;          // B_N * D_K
  float* pn   = fn + B_N * D_K;        // B_N * D_K
  float* pos  = pn + B_N * D_K;        // B_N
  float* acc1 = pos + B_N;             // B_N
  float* acc2 = acc1 + B_N;            // B_N
  float* out  = (float*)d_out;

  ssc_normalize<<<2 * B_N, 128, 0, stream>>>(f, fpos, fn, pn, acc1, acc2);
  ssc_gram<<<TOTAL_WAVES / 8, 256, 0, stream>>>(fn, pn, pos, acc1, acc2);
  ssc_loss<<<1, 256, 0, stream>>>(pos, acc1, acc2, out);
}